// LCC_82764019794216
// MI455X (gfx1250) — compile-verified
//
#include <hip/hip_runtime.h>
#include <hip/hip_bf16.h>

// LCC (local cross-correlation) with 5x5x5 box window over (160,192,160) f32.
// - Halo-tile staging: GLOBAL_LOAD_ASYNC_TO_LDS_B32 per-lane gather (ASYNCcnt).
// - Box filters in x/y: banded-ones matmuls on V_WMMA_F32_16X16X4_F32.

typedef __attribute__((ext_vector_type(2))) float v2f;
typedef __attribute__((ext_vector_type(8))) float v8f;

#define DIMZ 160
#define DIMY 192
#define DIMX 160
#define STRY 160
#define STRZ (160 * 192)
#define NBLOCKS (10 * 12 * 160)

#if __has_builtin(__builtin_amdgcn_global_load_async_to_lds_b32)
#define HAVE_ASYNC_LDS 1
#endif

#define AS1 __attribute__((address_space(1)))
#define AS3 __attribute__((address_space(3)))

__device__ __forceinline__ int clampi(int v, int lo, int hi) {
    return v < lo ? lo : (v > hi ? hi : v);
}

// One f32 element: global (per-lane clamped gather address) -> LDS, async.
__device__ __forceinline__ void async_g2l(const float* g, float* l) {
#if defined(HAVE_ASYNC_LDS)
    __builtin_amdgcn_global_load_async_to_lds_b32((AS1 int*)g, (AS3 int*)l, 0, 0);
#else
    *l = *g;  // synchronous fallback
#endif
}

__device__ __forceinline__ void wait_async_lds() {
#if defined(HAVE_ASYNC_LDS)
#if __has_builtin(__builtin_amdgcn_s_wait_asynccnt)
    __builtin_amdgcn_s_wait_asynccnt(0);
#else
    asm volatile("s_wait_asynccnt 0x0" ::: "memory");
#endif
#endif
}

__global__ __launch_bounds__(128) void lcc_main_kernel(const float* __restrict__ F,
                                                       const float* __restrict__ M,
                                                       float* __restrict__ ws) {
    // P: raw 5-plane z-window of f and m (20x20 halo tiles), filled by async DMA
    // S: z-summed quantities over the 20x20 halo tile (y-local, x-local)
    // T: x-filtered (20 rows y, 16 cols x)
    // O: fully box-filtered 16x16 outputs
    __shared__ float P[2][5][400];
    __shared__ float S[5][20][20];
    __shared__ float T[5][20][16];
    __shared__ float O[5][16][16];
    __shared__ float wred[4];

    const int tid  = threadIdx.x;          // 0..127
    const int wave = tid >> 5;             // 0..3
    const int lane = tid & 31;             // wave32
    const int lhalf = lane & 15;           // 0..15
    const int hi = lane >> 4;              // 0 or 1 (half-wave)

    const int tx = blockIdx.x;             // 0..9
    const int ty = blockIdx.y;             // 0..11
    const int z  = blockIdx.z;             // 0..159
    const int x0 = tx * 16 - 2;
    const int y0 = ty * 16 - 2;

    // ---------- Phase 1a: async-gather the 2x5x(20x20) halo window into LDS ----------
    for (int e = tid; e < 4000; e += 128) {
        const int a   = (e >= 2000) ? 1 : 0;     // 0 = fixed, 1 = moving
        const int r   = e - a * 2000;            // 0..1999
        const int dz  = r / 400;                 // 0..4
        const int pos = r - dz * 400;            // 0..399
        const int i   = pos / 20;                // y-local
        const int j   = pos - i * 20;            // x-local
        const int zc  = clampi(z - 2 + dz, 0, DIMZ - 1);
        const int gy  = clampi(y0 + i, 0, DIMY - 1);
        const int gx  = clampi(x0 + j, 0, DIMX - 1);
        const size_t g = (size_t)zc * STRZ + (size_t)gy * STRY + (size_t)gx;
        const float* src = (a ? M : F) + g;
        async_g2l(src, &P[a][dz][pos]);
    }
    wait_async_lds();
    __syncthreads();

    // ---------- Phase 1b: z-window sums of the 5 quantities ----------
    for (int k = 0; k < 4; ++k) {
        const int pos = tid + 128 * k;
        if (pos < 400) {
            float sf = 0.f, sm = 0.f, sfm = 0.f, sff = 0.f, smm = 0.f;
            for (int dz = 0; dz < 5; ++dz) {
                const float f = P[0][dz][pos];
                const float m = P[1][dz][pos];
                sf  += f;
                sm  += m;
                sfm += f * m;
                sff += f * f;
                smm += m * m;
            }
            const int i = pos / 20;
            const int j = pos - i * 20;
            S[0][i][j] = sf;
            S[1][i][j] = sm;
            S[2][i][j] = sfm;
            S[3][i][j] = sff;
            S[4][i][j] = smm;
        }
    }
    __syncthreads();

    // ---------- Phase 2: x-direction box filter via WMMA ----------
    // T(20x16) = S(20x20) * Wx(20x16), Wx[k][n] = (n <= k <= n+4)
    // Two overlapping 16-row chunks (rows 0..15 and 4..19) cover the 20 rows.
    for (int t = wave; t < 10; t += 4) {            // wave-uniform loop
        const int q  = t >> 1;                      // quantity 0..4
        const int r0 = (t & 1) * 4;                 // row-chunk start: 0 or 4
        v8f acc = {};
        for (int c = 0; c < 5; ++c) {               // K chunks of 4 over x (20)
            // A fragment: rows r0+M, K = x cols; lane<16 -> K=4c,4c+1 ; lane>=16 -> 4c+2,4c+3
            const int k0 = 4 * c + 2 * hi;
            v2f a;
            a.x = S[q][r0 + lhalf][k0 + 0];
            a.y = S[q][r0 + lhalf][k0 + 1];
            // B fragment: banded ones, rows K, cols N=lhalf
            const int K0 = 4 * c + hi;              // v0: rows 4c (lo half) / 4c+1 (hi half)
            const int K1 = 4 * c + 2 + hi;          // v1: rows 4c+2 / 4c+3
            const int N  = lhalf;
            v2f b;
            b.x = (K0 >= N && K0 <= N + 4) ? 1.0f : 0.0f;
            b.y = (K1 >= N && K1 <= N + 4) ? 1.0f : 0.0f;
            acc = __builtin_amdgcn_wmma_f32_16x16x4_f32(
                false, a, false, b, (short)0, acc, false, false);
        }
        // D layout: VGPR vi -> row M=vi (lanes 0-15) / M=vi+8 (lanes 16-31), col N
        for (int vi = 0; vi < 8; ++vi) {
            T[q][r0 + vi + 8 * hi][lhalf] = acc[vi];
        }
    }
    __syncthreads();

    // ---------- Phase 3: y-direction box filter via WMMA ----------
    // O(16x16) = Wy(16x20) * T(20x16), Wy[m][k] = (m <= k <= m+4)
    for (int q = wave; q < 5; q += 4) {             // wave-uniform loop
        v8f acc = {};
        for (int c = 0; c < 5; ++c) {
            // A fragment: banded ones, M = output y, K = tmp rows
            const int Mr = lhalf;
            const int k0 = 4 * c + 2 * hi;
            v2f a;
            a.x = (k0 + 0 >= Mr && k0 + 0 <= Mr + 4) ? 1.0f : 0.0f;
            a.y = (k0 + 1 >= Mr && k0 + 1 <= Mr + 4) ? 1.0f : 0.0f;
            // B fragment: T rows 4c..4c+3, cols N=x
            v2f b;
            b.x = T[q][4 * c + hi][lhalf];
            b.y = T[q][4 * c + 2 + hi][lhalf];
            acc = __builtin_amdgcn_wmma_f32_16x16x4_f32(
                false, a, false, b, (short)0, acc, false, false);
        }
        for (int vi = 0; vi < 8; ++vi) {
            O[q][vi + 8 * hi][lhalf] = acc[vi];
        }
    }
    __syncthreads();

    // ---------- Phase 4: per-voxel LCC + block reduction ----------
    const float inv = 1.0f / 125.0f;
    float part = 0.f;
    for (int p = tid; p < 256; p += 128) {
        const int yy = p >> 4;
        const int xx = p & 15;
        const float Sf  = O[0][yy][xx];
        const float Sm  = O[1][yy][xx];
        const float Sfm = O[2][yy][xx];
        const float Sff = O[3][yy][xx];
        const float Smm = O[4][yy][xx];
        const float cross = Sfm - Sf * Sm * inv;
        const float fvar  = Sff - Sf * Sf * inv;
        const float mvar  = Smm - Sm * Sm * inv;
        part += cross * cross / (fvar * mvar + 0.1f);
    }
    // wave32 shuffle reduction, then across the 4 waves
    for (int o = 16; o > 0; o >>= 1) part += __shfl_down(part, o, 32);
    if (lane == 0) wred[wave] = part;
    __syncthreads();
    if (tid == 0) {
        const int bflat = (blockIdx.z * gridDim.y + blockIdx.y) * gridDim.x + blockIdx.x;
        ws[bflat] = wred[0] + wred[1] + wred[2] + wred[3];
    }
}

// Deterministic final reduction: single block, fixed traversal order.
__global__ __launch_bounds__(256) void lcc_reduce_kernel(const float* __restrict__ ws,
                                                         float* __restrict__ out,
                                                         int n) {
    __shared__ float sh[256];
    float s = 0.f;
    for (int i = threadIdx.x; i < n; i += 256) s += ws[i];
    sh[threadIdx.x] = s;
    __syncthreads();
    for (int o = 128; o > 0; o >>= 1) {
        if (threadIdx.x < o) sh[threadIdx.x] += sh[threadIdx.x + o];
        __syncthreads();
    }
    if (threadIdx.x == 0) out[0] = -sh[0];
}

extern "C" void kernel_launch(void* const* d_in, const int* in_sizes, int n_in,
                              void* d_out, int out_size, void* d_ws, size_t ws_size,
                              hipStream_t stream) {
    const float* F = (const float*)d_in[0];   // im_fixed
    const float* M = (const float*)d_in[1];   // im_moving_warped
    float* ws  = (float*)d_ws;                // NBLOCKS partials (76.8 KB)
    float* out = (float*)d_out;

    dim3 grid(10, 12, 160);                   // x-tiles, y-tiles, z
    lcc_main_kernel<<<grid, 128, 0, stream>>>(F, M, ws);
    lcc_reduce_kernel<<<1, 256, 0, stream>>>(ws, out, NBLOCKS);
}